// Coords2TypedCoords_58841051955820
// MI455X (gfx1250) — compile-verified
//
#include <hip/hip_runtime.h>
#include <hip/hip_bf16.h>
#include <stdint.h>

#ifndef __has_builtin
#define __has_builtin(x) 0
#endif

#if __has_builtin(__builtin_amdgcn_global_store_async_from_lds_b128) && \
    __has_builtin(__builtin_amdgcn_s_wait_asynccnt)
#define USE_ASYNC_STORE 1
#else
#define USE_ASYNC_STORE 0
#endif

#define AS_GLOBAL __attribute__((address_space(1)))
#define AS_LDS    __attribute__((address_space(3)))

typedef int v4i __attribute__((vector_size(16)));

namespace {
constexpr int kB    = 1024;          // batch rows
constexpr int kN    = 8192;          // atoms per row
constexpr int kT    = 11;            // atom types
constexpr int kNB   = kT + 1;        // buckets incl. invalid (type 11)
constexpr int kNT   = 256;           // threads per block (8 waves, wave32)
constexpr int kAPT  = kN / kNT;      // 32 atoms per thread (contiguous -> stable)
constexpr int kN3   = 3 * kN;        // 24576 floats per output row
constexpr int kHalf = kN / 2;        // 4096 atoms staged per LDS pass
constexpr int kHalfF = kHalf * 3;    // 12288 floats = 48 KB
constexpr int kVecPerThread = kHalfF / 4 / kNT; // 12 float4 stores per thread per half
}

__global__ __launch_bounds__(kNT) void coords2typed_kernel(
    const float* __restrict__ coords, const int* __restrict__ types,
    const int* __restrict__ nums, float* __restrict__ out)
{
  __shared__ float s_half[kHalfF];     // 48 KB: staged half of the sorted row
  __shared__ int   s_hist[kNB * kNT];  // 12 KB: flat hist -> exclusive-scan bases
  __shared__ int   s_sums[kNT];        // 1 KB : per-thread chunk sums

  const int b      = blockIdx.x;
  const int tid    = threadIdx.x;
  const int na     = nums[b];
  const int base_i = tid * kAPT;

  // ---- load this thread's 32 contiguous atom types (int4 = 16B loads)
  const int* trow = types + (size_t)b * kN + base_i;
  int ty[kAPT];
  #pragma unroll
  for (int k = 0; k < kAPT; k += 4) {
    int4 v = *reinterpret_cast<const int4*>(trow + k);
    ty[k] = v.x; ty[k + 1] = v.y; ty[k + 2] = v.z; ty[k + 3] = v.w;
  }
  #pragma unroll
  for (int k = 0; k < kAPT; ++k)
    if (base_i + k >= na) ty[k] = kT;   // invalid atoms -> bucket 11

  // ---- packed histogram + stable within-thread rank
  // 12 buckets x 6-bit counters: types 0..9 in a u64, types 10..11 in a u32.
  // rank[k] = counter value before increment  (stable order within thread)
  unsigned long long h0 = 0ull;
  unsigned int       h1 = 0u;
  int rnk[kAPT];
  #pragma unroll
  for (int k = 0; k < kAPT; ++k) {
    int t = ty[k];
    int sh  = 6 * t;                 // 0..66 (only valid for t<10 on h0)
    int sh1 = (sh - 60) & 31;        // 0 or 6 for t=10/11 (garbage otherwise, masked off)
    bool lo = t < 10;
    int r0 = (int)((h0 >> (sh & 63)) & 63ull);
    int r1 = (int)((h1 >> sh1) & 63u);
    rnk[k] = lo ? r0 : r1;
    h0 += lo ? (1ull << (sh & 63)) : 0ull;
    h1 += lo ? 0u : (1u << sh1);
  }
  // unpack per-type counts (static shifts)
  int h[kNB];
  #pragma unroll
  for (int c = 0; c < 10; ++c) h[c] = (int)((h0 >> (6 * c)) & 63ull);
  h[10] = (int)(h1 & 63u);
  h[11] = (int)((h1 >> 6) & 63u);

  // flat layout: f = type*256 + thread  (type-major => scan order == stable order)
  #pragma unroll
  for (int c = 0; c < kNB; ++c) s_hist[c * kNT + tid] = h[c];
  __syncthreads();

  // ---- exclusive scan over the flat 3072-entry table
  // each thread scans 12 consecutive flat entries
  int cv[kNB]; int csum = 0;
  #pragma unroll
  for (int c = 0; c < kNB; ++c) { cv[c] = s_hist[tid * kNB + c]; csum += cv[c]; }
  s_sums[tid] = csum;
  __syncthreads();
  // Hillis-Steele inclusive scan over 256 partial sums
  for (int off = 1; off < kNT; off <<= 1) {
    int v  = s_sums[tid];
    int vo = (tid >= off) ? s_sums[tid - off] : 0;
    __syncthreads();
    s_sums[tid] = v + vo;
    __syncthreads();
  }
  int run = (tid > 0) ? s_sums[tid - 1] : 0;
  #pragma unroll
  for (int c = 0; c < kNB; ++c) { s_hist[tid * kNB + c] = run; run += cv[c]; }
  __syncthreads();

  // counts / offsets outputs (reference returns int32; d_out is float -> cast)
  if (tid < kT) {
    int o0 = s_hist[tid * kNT];          // bucket start
    int o1 = s_hist[(tid + 1) * kNT];    // next bucket start (tid=10 -> invalid bucket)
    out[(size_t)kB * kN3 + (size_t)b * kT + tid] = (float)(o1 - o0);
    out[(size_t)kB * kN3 + (size_t)kB * kT + (size_t)b * kT + tid] = (float)o0;
  }

  // ---- stable destinations: one conflict-free dynamic LDS read per atom.
  // word index = t*256 + tid; 256 % 64banks == 0 -> bank = tid % 64, no conflicts.
  int dst[kAPT];
  #pragma unroll
  for (int k = 0; k < kAPT; ++k)
    dst[k] = s_hist[ty[k] * kNT + tid] + rnk[k];

  const float* crow = coords + (size_t)b * kN3 + (size_t)base_i * 3;
  float* orow = out + (size_t)b * kN3;

  // ---- scatter into LDS in two halves, stream each half out coalesced
  #pragma unroll
  for (int half = 0; half < 2; ++half) {
    const int lo = half * kHalf, hi = lo + kHalf;
    #pragma unroll
    for (int k = 0; k < kAPT; ++k) {
      int d = dst[k];
      if (d >= lo && d < hi) {
        float x = 0.f, y = 0.f, z = 0.f;
        if (ty[k] < kT) { x = crow[3 * k]; y = crow[3 * k + 1]; z = crow[3 * k + 2]; }
        int p = (d - lo) * 3;
        s_half[p] = x; s_half[p + 1] = y; s_half[p + 2] = z;
      }
    }
    __syncthreads();

    float* oh = orow + (size_t)lo * 3;
#if USE_ASYNC_STORE
    // gfx1250 async LDS->global DMA: full 128-bit per-lane stores, ASYNCcnt-tracked
    #pragma unroll
    for (int k = 0; k < kVecPerThread; ++k) {
      int idx = (k * kNT + tid) * 4;
      __builtin_amdgcn_global_store_async_from_lds_b128(
          (AS_GLOBAL v4i*)(oh + idx), (AS_LDS v4i*)(s_half + idx), 0, 0);
    }
    __builtin_amdgcn_s_wait_asynccnt(0);  // LDS half is reused next pass
#else
    #pragma unroll
    for (int k = 0; k < kVecPerThread; ++k) {
      int idx = (k * kNT + tid) * 4;
      float4 v = *reinterpret_cast<const float4*>(s_half + idx);
      *reinterpret_cast<float4*>(oh + idx) = v;
    }
#endif
    __syncthreads();
  }
}

extern "C" void kernel_launch(void* const* d_in, const int* in_sizes, int n_in,
                              void* d_out, int out_size, void* d_ws, size_t ws_size,
                              hipStream_t stream) {
  (void)in_sizes; (void)n_in; (void)out_size; (void)d_ws; (void)ws_size;
  const float* coords = (const float*)d_in[0];   // (B, 3N) float32
  const int*   types  = (const int*)d_in[1];     // (B, N) int32
  const int*   nums   = (const int*)d_in[2];     // (B,) int32
  float* out = (float*)d_out;                    // coords | counts | offsets (flat)

  coords2typed_kernel<<<kB, kNT, 0, stream>>>(coords, types, nums, out);
}